// RoiPoolingConv_3100966388019
// MI455X (gfx1250) — compile-verified
//
#include <hip/hip_runtime.h>

typedef __attribute__((ext_vector_type(2))) float v2f;
typedef __attribute__((ext_vector_type(8))) float v8f;

#define POOLP 14
#define KCOLS 24          // max staged source columns per ROI row (w<24 -> span <= 23)
#define CH 1024
#define CHALF 512
#define LDS_STRIDE 520    // 512 + 8 floats pad: bank = (k*8 + c) % 64 -> conflict-free B reads
#define IMG_W 64
#define IMG_H 64

__global__ __launch_bounds__(256)
void roi_pool_wmma_kernel(const float* __restrict__ img,
                          const int* __restrict__ rois,
                          float* __restrict__ out)
{
    __shared__ float sm[KCOLS * LDS_STRIDE];   // rowInterp[k][c_local], 49.9 KB

    const int b     = blockIdx.x;
    const int roi   = b / (POOLP * 2);
    const int rem   = b % (POOLP * 2);
    const int py    = rem >> 1;
    const int chalf = rem & 1;

    // uniform (SGPR) ROI fetch
    const int rx = rois[roi * 4 + 0];
    const int ry = rois[roi * 4 + 1];
    const int rw = rois[roi * 4 + 2];
    const int rh = rois[roi * 4 + 3];

    // vertical source coords for this output row (uniform) — mirrors reference math
    const float hs = (float)rh / 14.0f;
    const float ys = (float)ry + (float)py * hs;
    int y0 = (int)floorf(ys);
    y0 = min(max(y0, 0), IMG_H - 1);
    const int   y1 = min(y0 + 1, IMG_H - 1);
    const float wy = ys - (float)y0;

    const int tid    = threadIdx.x;
    const int cbaseg = chalf * CHALF;

    // ---------------- vertical lerp: global -> LDS ----------------
    // stage KCOLS source columns starting at rx (rx+23 <= 62, always in-bounds)
    const float* rowT = img + ((size_t)(y0 * IMG_W + rx) * CH) + cbaseg;
    const float* rowBm = img + ((size_t)(y1 * IMG_W + rx) * CH) + cbaseg;
    const float  omwy = 1.0f - wy;

    #pragma unroll
    for (int it = 0; it < 12; ++it) {          // 24 rows * 128 float4 / 256 thr
        int idx = it * 256 + tid;              // 0..3071
        int k   = idx >> 7;                    // 0..23
        int c4  = (idx & 127) << 2;            // 0..508
        const float4 a = *(const float4*)(rowT  + (size_t)k * CH + c4);
        const float4 c = *(const float4*)(rowBm + (size_t)k * CH + c4);
        float4 v;
        v.x = a.x * omwy + c.x * wy;
        v.y = a.y * omwy + c.y * wy;
        v.z = a.z * omwy + c.z * wy;
        v.w = a.w * omwy + c.w * wy;
        *(float4*)(&sm[k * LDS_STRIDE + c4]) = v;
    }
    __syncthreads();

    // ---------------- horizontal pass: WMMA f32 16x16x4 ----------------
    const int lane = tid & 31;
    const int wv   = tid >> 5;
    const int m    = lane & 15;                // output px row of A (rows 14,15 = pad, results discarded)
    const int kh   = (lane >> 4) << 1;         // lanes 0-15: K=0,1 ; lanes 16-31: K=2,3

    // per-output-column bilinear weights (reference math)
    const float wsx = (float)rw / 14.0f;
    const float xs  = (float)rx + (float)m * wsx;
    int x0 = (int)floorf(xs);
    x0 = min(max(x0, 0), IMG_W - 1);
    const float fx   = xs - (float)x0;
    const int   ox0  = x0 - rx;                // 0..22 (< KCOLS-1)
    const float omfx = 1.0f - fx;

    // A-matrix fragments: Wx[m][j], 2 nonzeros per row, built once per wave
    v2f aK[6];
    #pragma unroll
    for (int kk = 0; kk < 6; ++kk) {
        int j0 = kk * 4 + kh;
        int j1 = j0 + 1;
        float a0 = (j0 == ox0) ? omfx : ((j0 == ox0 + 1) ? fx : 0.0f);
        float a1 = (j1 == ox0) ? omfx : ((j1 == ox0 + 1) ? fx : 0.0f);
        aK[kk].x = a0;
        aK[kk].y = a1;
    }

    const int    n       = lane & 15;          // channel column within tile
    const int    pxhi    = (lane >> 4) << 3;   // 0 or 8
    const size_t outbase = (((size_t)roi * POOLP + py) * POOLP) * CH + cbaseg;

    #pragma unroll
    for (int t = 0; t < 4; ++t) {              // 32 channel tiles / 8 waves
        const int ct = wv * 4 + t;
        const int c0 = ct * 16 + n;            // this lane's local channel column

        v8f acc = (v8f){0.f, 0.f, 0.f, 0.f, 0.f, 0.f, 0.f, 0.f};
        #pragma unroll
        for (int kk = 0; kk < 6; ++kk) {
            const int kb = kk * 4 + kh;
            v2f bv;
            bv.x = sm[kb * LDS_STRIDE + c0];
            bv.y = sm[(kb + 1) * LDS_STRIDE + c0];
            acc = __builtin_amdgcn_wmma_f32_16x16x4_f32(
                    /*neg_a=*/false, aK[kk],
                    /*neg_b=*/false, bv,
                    /*c_mod=*/(short)0, acc,
                    /*reuse_a=*/false, /*reuse_b=*/false);
        }

        // D layout: VGPR r -> px = r (lanes 0-15) / r+8 (lanes 16-31); skip pad rows 14,15
        #pragma unroll
        for (int r = 0; r < 8; ++r) {
            const int px = r + pxhi;
            if (px < POOLP) {
                __builtin_nontemporal_store(
                    acc[r], out + outbase + (size_t)px * CH + c0);
            }
        }
    }
}

extern "C" void kernel_launch(void* const* d_in, const int* in_sizes, int n_in,
                              void* d_out, int out_size, void* d_ws, size_t ws_size,
                              hipStream_t stream) {
    const float* img  = (const float*)d_in[0];
    const int*   rois = (const int*)d_in[1];
    float*       out  = (float*)d_out;

    const int nrois  = in_sizes[1] / 4;        // 300
    const int blocks = nrois * POOLP * 2;      // (roi, py, channel-half)

    roi_pool_wmma_kernel<<<blocks, 256, 0, stream>>>(img, rois, out);
}